// ModelNew_3556232922061
// MI455X (gfx1250) — compile-verified
//
#include <hip/hip_runtime.h>

typedef __attribute__((ext_vector_type(2))) float v2f;
typedef __attribute__((ext_vector_type(8))) float v8f;

#define CIN   64
#define COUT  128
#define HH    128
#define WW    128
#define NTAPS 9
#define KTOT  (CIN * NTAPS)      // 576
#define WLDS_STRIDE 580          // 576 + 4 pad: 580 % 64 == 4 -> conflict-free A b64 loads
#define ACT_IC_PAD 68            // ic dim padded 64->68: (4c+ic)%64 -> conflict-free B b64 loads
#define ACT_C_COUNT 34
#define TILE_W 32
#define EPSV 1e-5f

// ---- pack w (ic, oc, kh, kw) -> wpk[oc][tap][ic], tap = kh*3+kw ----------
__global__ void pack_w_kernel(const float* __restrict__ w, float* __restrict__ wpk) {
    int idx = blockIdx.x * blockDim.x + threadIdx.x;
    if (idx >= COUT * KTOT) return;
    int ic  = idx & (CIN - 1);
    int t   = idx >> 6;           // = oc*9 + tap
    int tap = t % NTAPS;
    int oc  = t / NTAPS;
    wpk[oc * KTOT + tap * CIN + ic] = w[(ic * COUT + oc) * NTAPS + tap];
}

__device__ __forceinline__ float gelu_tanh(float y) {
    float u  = 0.7978845608f * (y + 0.044715f * y * y * y);
    float e  = __expf(2.0f * u);          // tanh(u) = 1 - 2/(e^{2u}+1); correct limits at +-inf
    float th = 1.0f - 2.0f / (e + 1.0f);
    return 0.5f * y * (1.0f + th);
}

// ---- fused convT(crop) + GELU + per-pixel groupnorm + affine -------------
__global__ __launch_bounds__(256) void conv_gelu_gn_kernel(
    const float* __restrict__ x,    const float* __restrict__ wpk,
    const float* __restrict__ bias, const float* __restrict__ gnw,
    const float* __restrict__ gnb,  float* __restrict__ out) {

    __shared__ float wlds[COUT * WLDS_STRIDE];              // 296,960 B
    __shared__ float act[3 * ACT_C_COUNT * ACT_IC_PAD];     //  27,744 B  [kh][c][ic]

    const int tid  = threadIdx.x;
    const int lane = tid & 31;
    const int wave = tid >> 5;          // 8 waves; wave == groupnorm group id
    const int hi   = lane >> 4;         // lane half
    const int pl   = lane & 15;         // pixel within 16-tile / A row
    const int koff = hi * 2;            // k sub-index held by this half (A & B consistent)

    const int bx   = blockIdx.x;        // = nimg*128 + oh
    const int nimg = bx >> 7;
    const int oh   = bx & (HH - 1);

    // stage all 128x576 f32 weights into padded LDS (once per block, L2-resident source)
    {
        const float4* src = (const float4*)wpk;
        for (int e = tid; e < COUT * KTOT / 4; e += 256) {
            int oc = e / (KTOT / 4);
            int k4 = e - oc * (KTOT / 4);
            float4 v = src[e];
            float* dst = &wlds[oc * WLDS_STRIDE + k4 * 4];
            dst[0] = v.x; dst[1] = v.y; dst[2] = v.z; dst[3] = v.w;
        }
    }

    // per-wave channel params: element (vgpr r, lane) has oc = ocb + r + 8*hi
    const int ocb = wave * 16;
    float bb[8], gw[8], gb[8];
#pragma unroll
    for (int r = 0; r < 8; ++r) {
        int oc = ocb + r + 8 * hi;
        bb[r] = bias[oc]; gw[r] = gnw[oc]; gb[r] = gnb[oc];
    }

    const unsigned arow = (unsigned)(ocb + pl) * WLDS_STRIDE + koff;

    __syncthreads();

    for (int ow0 = 0; ow0 < WW; ow0 += TILE_W) {
        // ---- stage activation halo tile: act[kh][c][ic], c = xw - (ow0-2) ----
        // order: c fastest -> coalesced global reads (34*4B runs)
        for (int e = tid; e < CIN * 3 * ACT_C_COUNT; e += 256) {
            int c   = e % ACT_C_COUNT;
            int rem = e / ACT_C_COUNT;
            int kh  = rem % 3;
            int ic  = rem / 3;
            int xr  = oh - kh;
            int xc  = ow0 - 2 + c;
            float v = 0.0f;
            if (xr >= 0 && xc >= 0)
                v = x[(((nimg * CIN + ic) * HH) + xr) * WW + xc];
            act[(kh * ACT_C_COUNT + c) * ACT_IC_PAD + ic] = v;
        }
        __syncthreads();

        // ---- implicit GEMM over K = 9 taps x 64 ic, f32 WMMA 16x16x4 ----
        v8f acc0 = {};   // pixels ow0 + 0..15
        v8f acc1 = {};   // pixels ow0 + 16..31
        for (int tap = 0; tap < NTAPS; ++tap) {
            const int kh = tap / 3;
            const int kw = tap - kh * 3;
            const float* ab  = &wlds[arow + tap * CIN];
            const float* bp0 = &act[(kh * ACT_C_COUNT + (pl + 2 - kw)) * ACT_IC_PAD + koff];
            const float* bp1 = bp0 + 16 * ACT_IC_PAD;   // c += 16 -> second pixel half-tile
#pragma unroll
            for (int icb = 0; icb < CIN; icb += 4) {
                v2f a  = *(const v2f*)(ab  + icb);   // ds_load_b64 (banks 4m..4m+3)
                v2f b0 = *(const v2f*)(bp0 + icb);   // ds_load_b64 (banks 4c+{0..3})
                v2f b1 = *(const v2f*)(bp1 + icb);
                acc0 = __builtin_amdgcn_wmma_f32_16x16x4_f32(
                           false, a, false, b0, (short)0, acc0, false, false);
                acc1 = __builtin_amdgcn_wmma_f32_16x16x4_f32(
                           false, a, false, b1, (short)0, acc1, false, false);
            }
        }

        // ---- epilogue: bias + GELU + per-pixel group stats (wave-local) ----
        float g0[8], g1[8];
        float s1a = 0.f, s2a = 0.f, s1b = 0.f, s2b = 0.f;
#pragma unroll
        for (int r = 0; r < 8; ++r) {
            float ya = acc0[r] + bb[r];
            float yb = acc1[r] + bb[r];
            float ga  = gelu_tanh(ya);
            float gbv = gelu_tanh(yb);
            g0[r] = ga; g1[r] = gbv;
            s1a += ga;  s2a += ga * ga;
            s1b += gbv; s2b += gbv * gbv;
        }
        // combine the two lane-halves (channels r and r+8 of the same pixel)
        s1a += __shfl_xor(s1a, 16, 32);  s2a += __shfl_xor(s2a, 16, 32);
        s1b += __shfl_xor(s1b, 16, 32);  s2b += __shfl_xor(s2b, 16, 32);
        float ma = s1a * (1.0f / 16.0f);
        float mb = s1b * (1.0f / 16.0f);
        float va = s2a * (1.0f / 16.0f) - ma * ma;
        float vb = s2b * (1.0f / 16.0f) - mb * mb;
        float ia = rsqrtf(va + EPSV);
        float ib = rsqrtf(vb + EPSV);

#pragma unroll
        for (int r = 0; r < 8; ++r) {
            int oc = ocb + r + 8 * hi;
            long obase = (((long)nimg * COUT + oc) * HH + oh) * WW;
            out[obase + ow0 + pl]      = (g0[r] - ma) * ia * gw[r] + gb[r];
            out[obase + ow0 + 16 + pl] = (g1[r] - mb) * ib * gw[r] + gb[r];
        }
        __syncthreads();   // protect act before next tile's staging
    }
}

extern "C" void kernel_launch(void* const* d_in, const int* in_sizes, int n_in,
                              void* d_out, int out_size, void* d_ws, size_t ws_size,
                              hipStream_t stream) {
    const float* x   = (const float*)d_in[0];
    const float* w   = (const float*)d_in[1];
    const float* b   = (const float*)d_in[2];
    const float* gnw = (const float*)d_in[3];
    const float* gnb = (const float*)d_in[4];
    float* out = (float*)d_out;
    float* wpk = (float*)d_ws;   // 128*576*4 = 294,912 B

    pack_w_kernel<<<(COUT * KTOT + 255) / 256, 256, 0, stream>>>(w, wpk);

    const int nblocks = 32 * HH;   // one (n, oh) row per block
    conv_gelu_gn_kernel<<<nblocks, 256, 0, stream>>>(x, wpk, b, gnw, gnb, out);
}